// NAG_13907104105204
// MI455X (gfx1250) — compile-verified
//
#include <hip/hip_runtime.h>

// NAG solver: x_{t+1} = y_t - s*(w@y_t - b); y_{t+1} = (1+a)x_{t+1} - a*x_t
// One block per batch (4096 blocks x 256 threads = 8 waves).
// Each wave owns a 16-row block of w, kept resident in VGPRs as WMMA B-tiles.
// y (128 floats) lives in LDS and is rebuilt each iteration.

typedef __attribute__((ext_vector_type(2))) float v2f;
typedef __attribute__((ext_vector_type(8))) float v8f;

#define K_ITERS 20
#define MU_CONST 0.05f
#define NB 4096
#define NN 128

__global__ __launch_bounds__(256)
void nag_wmma_kernel(const float* __restrict__ w,
                     const float* __restrict__ b,
                     const float* __restrict__ sp,
                     float* __restrict__ out)
{
    __shared__ float sy[NN];

    const int tid   = threadIdx.x;
    const int lane  = tid & 31;
    const int wave  = tid >> 5;
    const int n     = lane & 15;   // row within this wave's 16-row block (B-matrix N index)
    const int half  = lane >> 4;   // lane-half selects which K pair (K=0,1 vs K=2,3)
    const int rb    = wave << 4;   // first row of the 16-row block
    const int batch = blockIdx.x;

    const float s     = *sp;
    const float sqms  = sqrtf(MU_CONST * s);
    const float alpha = (1.0f - sqms) / (1.0f + sqms);

    // ---- one-time streaming load of this wave's 16x128 w-block as 32 B-tiles ----
    // B-tile kc holds B[k,n] = w[rb+n, 4*kc + k]; lane n: (k=0,1), lane n+16: (k=2,3).
    const float* wp = w + (size_t)batch * (NN * NN) + (size_t)(rb + n) * NN + half * 2;
    v2f wt[32];
#pragma unroll
    for (int kc = 0; kc < 32; ++kc)
        wt[kc] = *(const v2f*)(wp + kc * 4);

    const float bv = b[(size_t)batch * NN + rb + n];

    if (tid < NN) sy[tid] = 0.0f;   // y0 = 0
    __syncthreads();

    float xl = 0.0f, yl = 0.0f;     // this lane's x,y component (row rb+n), dup per half

#pragma unroll 1
    for (int it = 0; it < K_ITERS; ++it) {
        v8f acc0 = {}, acc1 = {}, acc2 = {}, acc3 = {};
#pragma unroll
        for (int kc = 0; kc < 32; kc += 4) {
            // A-tile: A[m,k] = y[4*kc + k] broadcast over all m rows.
            // Per 32-bit 16x4 A layout: lanes0-15 get (K0,K1), lanes16-31 get (K2,K3).
            v2f a0 = *(const v2f*)(&sy[(kc + 0) * 4 + half * 2]);
            v2f a1 = *(const v2f*)(&sy[(kc + 1) * 4 + half * 2]);
            v2f a2 = *(const v2f*)(&sy[(kc + 2) * 4 + half * 2]);
            v2f a3 = *(const v2f*)(&sy[(kc + 3) * 4 + half * 2]);
            acc0 = __builtin_amdgcn_wmma_f32_16x16x4_f32(false, a0, false, wt[kc + 0],
                                                         (short)0, acc0, false, false);
            acc1 = __builtin_amdgcn_wmma_f32_16x16x4_f32(false, a1, false, wt[kc + 1],
                                                         (short)0, acc1, false, false);
            acc2 = __builtin_amdgcn_wmma_f32_16x16x4_f32(false, a2, false, wt[kc + 2],
                                                         (short)0, acc2, false, false);
            acc3 = __builtin_amdgcn_wmma_f32_16x16x4_f32(false, a3, false, wt[kc + 3],
                                                         (short)0, acc3, false, false);
        }
        // All D rows identical (A rows broadcast) -> VGPR0 of each acc holds
        // grad partial for row rb + (lane&15) in every lane.
        float g  = (acc0[0] + acc1[0]) + (acc2[0] + acc3[0]);
        float xn = yl - s * (g - bv);
        float yn = (1.0f + alpha) * xn - alpha * xl;
        xl = xn;
        yl = yn;

        __syncthreads();                 // all waves finished reading sy this iter
        if (!(lane & 16)) sy[rb + n] = yn;
        __syncthreads();                 // new y visible to every wave
    }

    if (!(lane & 16)) {
        out[(size_t)batch * NN + rb + n] = xl;                        // x
        out[(size_t)NB * NN + (size_t)batch * NN + rb + n] = yl;      // y
    }
}

extern "C" void kernel_launch(void* const* d_in, const int* in_sizes, int n_in,
                              void* d_out, int out_size, void* d_ws, size_t ws_size,
                              hipStream_t stream) {
    (void)in_sizes; (void)n_in; (void)d_ws; (void)ws_size; (void)out_size;
    const float* w = (const float*)d_in[0];
    const float* b = (const float*)d_in[1];
    const float* s = (const float*)d_in[2];
    float* out = (float*)d_out;
    nag_wmma_kernel<<<dim3(NB), dim3(256), 0, stream>>>(w, b, s, out);
}